// GraphEncoder_24489903521882
// MI455X (gfx1250) — compile-verified
//
#include <hip/hip_runtime.h>
#include <hip/hip_bf16.h>
#include <stdint.h>

#define N_NODES 100000
#define N_EDGES 1600000
#define D_IN 128
#define HDIM 64

typedef __attribute__((ext_vector_type(2))) float v2f;
typedef __attribute__((ext_vector_type(8))) float v8f;

#define WAVES 4
#define A_STRIDE 132   // 128 + 4 pad -> LDS bank stride 4 (conflict-free col reads)
#define T_STRIDE 68    // 64 + 4 pad

// ---- CDNA5 async global->LDS copy (ASYNCcnt-tracked), 16B per lane ----
#define ASYNC_B128(ldsoff, gaddr, OFF)                                      \
    asm volatile("global_load_async_to_lds_b128 %0, %1, off offset:" #OFF   \
                 :: "v"(ldsoff), "v"(gaddr) : "memory")

__device__ __forceinline__ void wait_async0() {
    asm volatile("s_wait_asynccnt 0x0" ::: "memory");
}

__device__ __forceinline__ unsigned lds_off32(const void* p) {
    // generic pointer to LDS: low 32 bits are the LDS byte offset
    return (unsigned)(uintptr_t)p;
}

// native device-scope float max atomic (IEEE maximumNumber semantics)
__device__ __forceinline__ void atomicMaxF32(float* addr, float val) {
    asm volatile("global_atomic_max_num_f32 %0, %1, off scope:SCOPE_DEV"
                 :: "v"(addr), "v"(val) : "memory");
}

// One wave computes a 16-row x 64-col tile of  O = relu(A @ w1 + b1) @ w2 + b2.
// A: [16][128] in LDS (row stride A_STRIDE floats).
// If DIFF: LDS holds [x_dst | x_src]; logical A is [x_dst | x_src - x_dst],
// the subtraction is done while reading upper-K fragments.
// Weights are K-pair-interleaved: wpk[(k>>1)*64 + n] = {w[k][n], w[k+1][n]} (v2f).
// A and B fragments are double-buffered in source so the next k-step's loads
// overlap the current step's 4 WMMAs (no load->wait->wmma lock-step).
// Result in acc2[nt][r], WMMA f32 D layout: row=(lane>>4)*8+r, col=nt*16+(lane&15).
template <bool DIFF>
__device__ __forceinline__ void mlp2_wave_tile(
    const float* ldsA, float* ldsT,
    const v2f* __restrict__ w1p, const float* __restrict__ b1,
    const v2f* __restrict__ w2p, const float* __restrict__ b2,
    int lane, v8f acc2[4])
{
    const int col = lane & 15;
    const int kh  = (lane >> 4) & 1;

    auto ldA1 = [&](int k0) -> v2f {
        v2f a = *(const v2f*)(ldsA + col * A_STRIDE + k0 + 2 * kh);
        if (DIFF && k0 >= 64) {
            v2f base = *(const v2f*)(ldsA + col * A_STRIDE + (k0 - 64) + 2 * kh);
            a.x -= base.x;
            a.y -= base.y;
        }
        return a;
    };
    auto ldB1 = [&](int k0, v2f* o) {
#pragma unroll
        for (int nt = 0; nt < 4; ++nt)
            o[nt] = w1p[((k0 >> 1) + kh) * HDIM + nt * 16 + col];
    };

    // ---- stage 1: T = relu(A @ w1 + b1), K = 128 ----
    v8f acc[4];
#pragma unroll
    for (int nt = 0; nt < 4; ++nt) {
        float bv = b1[nt * 16 + col];
#pragma unroll
        for (int r = 0; r < 8; ++r) acc[nt][r] = bv;
    }
    {
        v2f bcur[4], bnxt[4];
        v2f acur, anxt;
        ldB1(0, bcur);
        acur = ldA1(0);
#pragma unroll
        for (int k0 = 0; k0 < 128; k0 += 4) {
            if (k0 + 4 < 128) {
                ldB1(k0 + 4, bnxt);
                anxt = ldA1(k0 + 4);
            }
#pragma unroll
            for (int nt = 0; nt < 4; ++nt)
                acc[nt] = __builtin_amdgcn_wmma_f32_16x16x4_f32(
                    false, acur, false, bcur[nt], (short)0, acc[nt], false, false);
            acur = anxt;
#pragma unroll
            for (int nt = 0; nt < 4; ++nt) bcur[nt] = bnxt[nt];
        }
    }
    // ReLU, then D-layout -> row-major LDS so it can be re-read in A-layout
#pragma unroll
    for (int nt = 0; nt < 4; ++nt) {
#pragma unroll
        for (int r = 0; r < 8; ++r) {
            float v = acc[nt][r];
            v = v > 0.0f ? v : 0.0f;
            ldsT[(kh * 8 + r) * T_STRIDE + nt * 16 + col] = v;
        }
    }

    // ---- stage 2: O = T @ w2 + b2, K = 64 ----
    auto ldA2 = [&](int k0) -> v2f {
        return *(const v2f*)(ldsT + col * T_STRIDE + k0 + 2 * kh);
    };
    auto ldB2 = [&](int k0, v2f* o) {
#pragma unroll
        for (int nt = 0; nt < 4; ++nt)
            o[nt] = w2p[((k0 >> 1) + kh) * HDIM + nt * 16 + col];
    };
#pragma unroll
    for (int nt = 0; nt < 4; ++nt) {
        float bv = b2[nt * 16 + col];
#pragma unroll
        for (int r = 0; r < 8; ++r) acc2[nt][r] = bv;
    }
    {
        v2f bcur[4], bnxt[4];
        v2f acur, anxt;
        ldB2(0, bcur);
        acur = ldA2(0);
#pragma unroll
        for (int k0 = 0; k0 < 64; k0 += 4) {
            if (k0 + 4 < 64) {
                ldB2(k0 + 4, bnxt);
                anxt = ldA2(k0 + 4);
            }
#pragma unroll
            for (int nt = 0; nt < 4; ++nt)
                acc2[nt] = __builtin_amdgcn_wmma_f32_16x16x4_f32(
                    false, acur, false, bcur[nt], (short)0, acc2[nt], false, false);
            acur = anxt;
#pragma unroll
            for (int nt = 0; nt < 4; ++nt) bcur[nt] = bnxt[nt];
        }
    }
}

// ---------------- node encoder: h = mlp2(x) ----------------
__global__ void __launch_bounds__(WAVES * 32)
encoder_kernel(const float* __restrict__ x,
               const v2f* __restrict__ w1p, const float* __restrict__ b1,
               const v2f* __restrict__ w2p, const float* __restrict__ b2,
               float* __restrict__ h)
{
    __shared__ float ldsA[WAVES][16 * A_STRIDE];
    __shared__ float ldsT[WAVES][16 * T_STRIDE];
    const int wave = threadIdx.x >> 5;
    const int lane = threadIdx.x & 31;
    const int tile = blockIdx.x * WAVES + wave;
    if (tile * 16 >= N_NODES) return;
    const int node0 = tile * 16;

    // async-stage A tile (16 x 128): 2 lanes per row, 64 floats (256B) each
    {
        const int r = lane >> 1, half = lane & 1;
        const float* g = x + (size_t)(node0 + r) * D_IN + half * 64;
        const unsigned l = lds_off32(ldsA[wave] + r * A_STRIDE + half * 64);
        ASYNC_B128(l, g, 0);   ASYNC_B128(l, g, 16);  ASYNC_B128(l, g, 32);
        ASYNC_B128(l, g, 48);  ASYNC_B128(l, g, 64);  ASYNC_B128(l, g, 80);
        ASYNC_B128(l, g, 96);  ASYNC_B128(l, g, 112); ASYNC_B128(l, g, 128);
        ASYNC_B128(l, g, 144); ASYNC_B128(l, g, 160); ASYNC_B128(l, g, 176);
        ASYNC_B128(l, g, 192); ASYNC_B128(l, g, 208); ASYNC_B128(l, g, 224);
        ASYNC_B128(l, g, 240);
    }
    wait_async0();

    v8f acc2[4];
    mlp2_wave_tile<false>(ldsA[wave], ldsT[wave], w1p, b1, w2p, b2, lane, acc2);

    const int col = lane & 15, kh = (lane >> 4) & 1;
#pragma unroll
    for (int nt = 0; nt < 4; ++nt)
#pragma unroll
        for (int r = 0; r < 8; ++r)
            h[(size_t)(node0 + kh * 8 + r) * HDIM + nt * 16 + col] = acc2[nt][r];
}

// ---------------- fused EdgeConv layer ----------------
__global__ void __launch_bounds__(WAVES * 32)
edge_kernel(const float* __restrict__ h,
            const int* __restrict__ srcI, const int* __restrict__ dstI,
            const v2f* __restrict__ w1p, const float* __restrict__ b1,
            const v2f* __restrict__ w2p, const float* __restrict__ b2,
            float* __restrict__ agg)
{
    __shared__ float ldsA[WAVES][16 * A_STRIDE];
    __shared__ float ldsT[WAVES][16 * T_STRIDE];
    const int wave = threadIdx.x >> 5;
    const int lane = threadIdx.x & 31;
    const int tile = blockIdx.x * WAVES + wave;
    if (tile * 16 >= N_EDGES) return;
    const int e0 = tile * 16;

    // async gather: x_dst -> A cols [0,64), raw x_src -> A cols [64,128)
    // 2 lanes per edge, 32 floats (128B) each half
    {
        const int r = lane >> 1, half = lane & 1;
        const int e = e0 + r;
        const int si = srcI[e];
        const int di = dstI[e];
        const float* gi = h + (size_t)di * HDIM + half * 32;
        const float* gj = h + (size_t)si * HDIM + half * 32;
        const unsigned li = lds_off32(ldsA[wave] + r * A_STRIDE + half * 32);
        const unsigned lj = li + 64 * 4;  // cols +64
        ASYNC_B128(li, gi, 0);  ASYNC_B128(li, gi, 16);
        ASYNC_B128(li, gi, 32); ASYNC_B128(li, gi, 48);
        ASYNC_B128(li, gi, 64); ASYNC_B128(li, gi, 80);
        ASYNC_B128(li, gi, 96); ASYNC_B128(li, gi, 112);
        ASYNC_B128(lj, gj, 0);  ASYNC_B128(lj, gj, 16);
        ASYNC_B128(lj, gj, 32); ASYNC_B128(lj, gj, 48);
        ASYNC_B128(lj, gj, 64); ASYNC_B128(lj, gj, 80);
        ASYNC_B128(lj, gj, 96); ASYNC_B128(lj, gj, 112);
    }
    wait_async0();

    v8f acc2[4];
    mlp2_wave_tile<true>(ldsA[wave], ldsT[wave], w1p, b1, w2p, b2, lane, acc2);

    // segment-max scatter to destination nodes (native max_num_f32 atomics)
    const int col = lane & 15, kh = (lane >> 4) & 1;
#pragma unroll
    for (int r = 0; r < 8; ++r) {
        const int node = dstI[e0 + kh * 8 + r];
#pragma unroll
        for (int nt = 0; nt < 4; ++nt)
            atomicMaxF32(&agg[(size_t)node * HDIM + nt * 16 + col], acc2[nt][r]);
    }
}

// ---------------- helpers ----------------
// Interleave K pairs: out[((k>>1)*64 + n)*2 + (k&1)] = w[k*64 + n]
__global__ void repack_pairs(const float* __restrict__ w, float* __restrict__ out, int total)
{
    int i = blockIdx.x * blockDim.x + threadIdx.x;
    if (i < total) {
        int k = i >> 6, n = i & 63;
        out[(((k >> 1) << 6) + n) * 2 + (k & 1)] = w[i];
    }
}

__global__ void fill_neg_inf(float4* __restrict__ p, int n4)
{
    int i = blockIdx.x * blockDim.x + threadIdx.x;
    if (i < n4) {
        float4 v;
        v.x = v.y = v.z = v.w = -__builtin_inff();
        p[i] = v;
    }
}

// agg==-inf (no incoming edges) -> 0, else relu(agg). Both equal max(agg, 0).
__global__ void finalize_relu(const float4* __restrict__ agg, float4* __restrict__ out, int n4)
{
    int i = blockIdx.x * blockDim.x + threadIdx.x;
    if (i < n4) {
        float4 v = agg[i];
        v.x = fmaxf(v.x, 0.0f); v.y = fmaxf(v.y, 0.0f);
        v.z = fmaxf(v.z, 0.0f); v.w = fmaxf(v.w, 0.0f);
        out[i] = v;
    }
}

extern "C" void kernel_launch(void* const* d_in, const int* in_sizes, int n_in,
                              void* d_out, int out_size, void* d_ws, size_t ws_size,
                              hipStream_t stream)
{
    const float* x    = (const float*)d_in[0];
    const int*   ei   = (const int*)d_in[1];
    const int*   srcI = ei;
    const int*   dstI = ei + N_EDGES;

    float* h    = (float*)d_ws;                   // [N_NODES][HDIM]
    float* agg  = h + (size_t)N_NODES * HDIM;     // [N_NODES][HDIM]
    float* wpk1 = agg + (size_t)N_NODES * HDIM;   // 128*64 floats (pair-interleaved w1)
    float* wpk2 = wpk1 + 128 * 64;                // 64*64 floats (pair-interleaved w2)

    const int enc_tiles  = N_NODES / 16;                       // 6250
    const int enc_blocks = (enc_tiles + WAVES - 1) / WAVES;    // 1563
    const int edge_tiles  = N_EDGES / 16;                      // 100000
    const int edge_blocks = (edge_tiles + WAVES - 1) / WAVES;  // 25000
    const int n4 = (N_NODES * HDIM) / 4;                       // 1.6M float4
    const int hb = (n4 + 255) / 256;

    // ---- encoder ----
    repack_pairs<<<32, 256, 0, stream>>>((const float*)d_in[2], wpk1, 128 * 64);
    repack_pairs<<<16, 256, 0, stream>>>((const float*)d_in[4], wpk2, 64 * 64);
    encoder_kernel<<<enc_blocks, WAVES * 32, 0, stream>>>(
        x, (const v2f*)wpk1, (const float*)d_in[3],
        (const v2f*)wpk2, (const float*)d_in[5], h);

    // ---- 3 EdgeConv layers ----
    for (int layer = 0; layer < 3; ++layer) {
        const float* w1 = (const float*)d_in[6 + layer * 4 + 0];
        const float* b1 = (const float*)d_in[6 + layer * 4 + 1];
        const float* w2 = (const float*)d_in[6 + layer * 4 + 2];
        const float* b2 = (const float*)d_in[6 + layer * 4 + 3];

        repack_pairs<<<32, 256, 0, stream>>>(w1, wpk1, 128 * 64);
        repack_pairs<<<16, 256, 0, stream>>>(w2, wpk2, 64 * 64);
        fill_neg_inf<<<hb, 256, 0, stream>>>((float4*)agg, n4);
        edge_kernel<<<edge_blocks, WAVES * 32, 0, stream>>>(
            h, srcI, dstI, (const v2f*)wpk1, b1, (const v2f*)wpk2, b2, agg);
        float* outp = (layer == 2) ? (float*)d_out : h;
        finalize_relu<<<hb, 256, 0, stream>>>((const float4*)agg, (float4*)outp, n4);
    }
}